// MinVarianceWeightsLayer_55396488184585
// MI455X (gfx1250) — compile-verified
//
#include <hip/hip_runtime.h>
#include <hip/hip_bf16.h>
#include <stdint.h>

typedef __attribute__((ext_vector_type(2))) float        v2f;
typedef __attribute__((ext_vector_type(8))) float        v8f;
typedef __attribute__((ext_vector_type(4))) float        f32x4;
typedef __attribute__((ext_vector_type(4))) unsigned int u32x4;
typedef __attribute__((ext_vector_type(8))) int          i32x8;
typedef __attribute__((ext_vector_type(4))) int          i32x4;

#define N   64
#define LD  65          // LDS row stride (floats): 64 + 1 pad -> conflict-free columns
#define NT  128         // 4 waves of 32

__global__ __launch_bounds__(NT)
void minvar_chol_kernel(const float* __restrict__ sigma, float* __restrict__ out)
{
    __shared__ float S[N * LD];

    const int tid  = threadIdx.x;
    const int lane = tid & 31;
    const int wid  = tid >> 5;
    const int b    = blockIdx.x;
    const float* gmat = sigma + (size_t)b * N * N;

    // ---------------------------------------------------------------
    // 1) Load 64x64 f32 matrix into LDS with row stride 65.
    //    CDNA5 path: single TDM DMA with LDS padding (1 DWORD per 64).
    // ---------------------------------------------------------------
#if defined(__gfx1250__) && __has_builtin(__builtin_amdgcn_tensor_load_to_lds)
    if (wid == 0) {
        unsigned long long ga = (unsigned long long)(uintptr_t)gmat;
        unsigned int lds_addr = (unsigned int)(uintptr_t)&S[0]; // low 32b of LDS aperture = LDS offset
        u32x4 g0;
        g0.x = 1u;                                   // count=1 valid descriptor, flags 0
        g0.y = lds_addr;                             // D#.lds_addr
        g0.z = (unsigned int)(ga & 0xffffffffu);     // D#.global_addr[31:0]
        g0.w = (unsigned int)(ga >> 32) | (2u << 30);// global_addr[56:32] | type=2 ("image")
        i32x8 g1;
        g1[0] = (2 << 16)      // data_size = 4 bytes
              | (1 << 20)      // pad_enable
              | (5 << 22);     // pad_interval code 5 = every 64 DWORDs; pad_amount code 0 = 1 DWORD
        g1[1] = (N << 16);     // tensor_dim0 low16 at bits[63:48] = 64
        g1[2] = (N << 16);     // tensor_dim0 hi16 = 0, tensor_dim1 low16 = 64
        g1[3] = (N << 16);     // tensor_dim1 hi16 = 0, tile_dim0 = 64
        g1[4] = N;             // tile_dim1 = 64, tile_dim2 = 0
        g1[5] = N;             // tensor_dim0_stride low32 = 64 elements
        g1[6] = 0;
        g1[7] = 0;
        i32x4 z4 = {0, 0, 0, 0};
        i32x8 z8 = {0, 0, 0, 0, 0, 0, 0, 0};
        __builtin_amdgcn_tensor_load_to_lds(g0, g1, z4, z4, z8, 0);
        __builtin_amdgcn_s_wait_tensorcnt(0);
    }
    __syncthreads();
#else
    for (int e = tid * 4; e < N * N; e += NT * 4) {
        f32x4 v = *(const f32x4*)(gmat + e);
        int r = e >> 6, c = e & 63;
        *(f32x4*)&S[r * LD + c] = v;
    }
    __syncthreads();
#endif

    // ---------------------------------------------------------------
    // 2) Symmetrize: S = 0.5*(S + S^T). Pair (i<j) owned by one thread.
    // ---------------------------------------------------------------
    for (int e = tid; e < N * N; e += NT) {
        int i = e >> 6, j = e & 63;
        if (j > i) {
            float m = 0.5f * (S[i * LD + j] + S[j * LD + i]);
            S[i * LD + j] = m;
            S[j * LD + i] = m;
        }
    }
    __syncthreads();

    // ---------------------------------------------------------------
    // 3) Blocked right-looking Cholesky, 4 panels of 16.
    // ---------------------------------------------------------------
    for (int p = 0; p < 4; ++p) {
        const int p0 = p * 16;

        // --- 3a) 16x16 diagonal Cholesky: wave 0, column-per-lane in VGPRs ---
        if (wid == 0) {
            float col[16];
            #pragma unroll
            for (int i = 0; i < 16; ++i) col[i] = 0.f;
            const int j = lane;
            if (j < 16) {
                #pragma unroll
                for (int i = 0; i < 16; ++i) col[i] = S[(p0 + i) * LD + p0 + j];
            }
            #pragma unroll
            for (int k = 0; k < 16; ++k) {
                float d     = __shfl(col[k], k);   // current A[k][k]
                float inv_s = rsqrtf(d);
                float mj    = 0.f;                 // will hold L[j][k]
                #pragma unroll
                for (int i = 0; i < 16; ++i) {
                    if (i >= k) {
                        float s = __shfl(col[i], k) * inv_s;  // L[i][k]
                        if (j == k) col[i] = s;               // lane k keeps scaled column
                        if (i == j) mj = s;
                        if (j > k && j < 16 && i >= j)        // A[i][j] -= L[i][k]*L[j][k]
                            col[i] -= s * mj;
                    }
                }
            }
            if (j < 16) {
                #pragma unroll
                for (int i = 0; i < 16; ++i)
                    if (i >= j) S[(p0 + i) * LD + p0 + j] = col[i];
            }
        }
        __syncthreads();

        if (p == 3) break;

        // --- 3b) Panel solve: L_ip = S_ip * Lpp^{-T}, one row per thread ---
        const int R = N - (p + 1) * 16;   // 48, 32, 16
        if (tid < R) {
            const int i = (p + 1) * 16 + tid;
            float r[16];
            #pragma unroll
            for (int c = 0; c < 16; ++c) r[c] = S[i * LD + p0 + c];
            #pragma unroll
            for (int c = 0; c < 16; ++c) {
                float acc = r[c];
                #pragma unroll
                for (int k = 0; k < 16; ++k)
                    if (k < c) acc -= r[k] * S[(p0 + c) * LD + p0 + k]; // Lpp[c][k], LDS broadcast
                r[c] = acc / S[(p0 + c) * LD + p0 + c];
            }
            #pragma unroll
            for (int c = 0; c < 16; ++c) S[i * LD + p0 + c] = r[c];
        }
        __syncthreads();

        // --- 3c) Trailing update: S_ij -= L_ip * L_jp^T via WMMA f32 16x16x4 ---
        {
            int t = 0;
            for (int jj = p + 1; jj < 4; ++jj) {
                for (int ii = jj; ii < 4; ++ii, ++t) {
                    if ((t & 3) != wid) continue;   // wave-uniform tile assignment
                    const int i0 = ii * 16, j0 = jj * 16;
                    const int lr   = lane & 15;
                    const int hi8  = (lane >= 16) ? 8 : 0;
                    const int ksel = (lane >= 16) ? 2 : 0;
#if defined(__gfx1250__)
                    v8f c;
                    #pragma unroll
                    for (int rr = 0; rr < 8; ++rr)
                        c[rr] = S[(i0 + rr + hi8) * LD + j0 + lr];
                    #pragma unroll
                    for (int ks = 0; ks < 16; ks += 4) {
                        v2f a, bf;
                        // A[m][k] = L[i0+m][p0+ks+k]; negate so D = C - A*B (f32 WMMA has no A-neg)
                        a[0]  = -S[(i0 + lr) * LD + p0 + ks + ksel + 0];
                        a[1]  = -S[(i0 + lr) * LD + p0 + ks + ksel + 1];
                        // B[k][n] = L_jp^T[k][n] = L[j0+n][p0+ks+k]
                        bf[0] =  S[(j0 + lr) * LD + p0 + ks + ksel + 0];
                        bf[1] =  S[(j0 + lr) * LD + p0 + ks + ksel + 1];
                        c = __builtin_amdgcn_wmma_f32_16x16x4_f32(
                                false, a, false, bf, (short)0, c, false, false);
                    }
                    #pragma unroll
                    for (int rr = 0; rr < 8; ++rr)
                        S[(i0 + rr + hi8) * LD + j0 + lr] = c[rr];
#else
                    // host-pass / non-gfx1250 scalar equivalent
                    for (int rr = 0; rr < 8; ++rr) {
                        int row = i0 + rr + hi8;
                        float acc = S[row * LD + j0 + lr];
                        for (int k = 0; k < 16; ++k)
                            acc -= S[row * LD + p0 + k] * S[(j0 + lr) * LD + p0 + k];
                        S[row * LD + j0 + lr] = acc;
                    }
#endif
                }
            }
        }
        __syncthreads();
    }

    // ---------------------------------------------------------------
    // 4) Solve L y = 1, L^T z = y (wave 0, two rows per lane), normalize.
    // ---------------------------------------------------------------
    if (wid == 0) {
        const int r0 = lane, r1 = lane + 32;
        float y0 = 1.f, y1 = 1.f;
        for (int k = 0; k < 64; ++k) {                  // forward substitution
            float dkk = S[k * LD + k];                  // LDS broadcast
            float yk  = ((k < 32) ? __shfl(y0, k) : __shfl(y1, k - 32)) / dkk;
            if (k < 32) { if (lane == k)      y0 = yk; }
            else        { if (lane == k - 32) y1 = yk; }
            if (r0 > k) y0 -= S[r0 * LD + k] * yk;      // column of L, conflict-free (LD=65)
            if (r1 > k) y1 -= S[r1 * LD + k] * yk;
        }
        float z0 = y0, z1 = y1;
        for (int k = 63; k >= 0; --k) {                 // backward substitution (L^T)
            float dkk = S[k * LD + k];
            float zk  = ((k < 32) ? __shfl(z0, k) : __shfl(z1, k - 32)) / dkk;
            if (k < 32) { if (lane == k)      z0 = zk; }
            else        { if (lane == k - 32) z1 = zk; }
            if (r0 < k) z0 -= S[k * LD + r0] * zk;      // L^T[i][k] = L[k][i], row access
            if (r1 < k) z1 -= S[k * LD + r1] * zk;
        }
        float s = z0 + z1;
        #pragma unroll
        for (int o = 16; o > 0; o >>= 1) s += __shfl_xor(s, o);
        float inv = 1.f / s;
        out[(size_t)b * N + r0] = z0 * inv;
        out[(size_t)b * N + r1] = z1 * inv;
    }
}

extern "C" void kernel_launch(void* const* d_in, const int* in_sizes, int n_in,
                              void* d_out, int out_size, void* d_ws, size_t ws_size,
                              hipStream_t stream)
{
    (void)n_in; (void)out_size; (void)d_ws; (void)ws_size;
    const float* sigma = (const float*)d_in[0];
    float* out = (float*)d_out;
    const int B = in_sizes[0] / (N * N);   // 8192
    minvar_chol_kernel<<<B, NT, 0, stream>>>(sigma, out);
}